// decode_81295140979425
// MI455X (gfx1250) — compile-verified
//
#include <hip/hip_runtime.h>
#include <hip/hip_bf16.h>
#include <stdint.h>

// ---------------- problem constants (match reference) ----------------
#define FH 512
#define FW 512
#define FC 80
#define NEL (FH * FW * FC)     // 20,971,520 scores
#define TOPK 100
#define BOX_SCALE 4.0f

// ---------------- pass-1 geometry ----------------
#define B1 512                  // blocks in pass 1
#define TPB1 256                // threads per block (8 waves, wave32)
#define VEC 4                   // float4 per lane per iteration
#define TILE (TPB1 * VEC)       // 1024 floats per block-iteration
#define CHUNK (NEL / B1)        // 40,960 elements per block
#define ITERS1 (CHUNK / TILE)   // 40 iterations
#define CAP1 2048               // LDS candidate buffer (power of 2)

// ---------------- pass-2 geometry ----------------
#define TPB2 1024
#define NCAND (B1 * TOPK)       // 51,200 candidates
#define ITERS2 (NCAND / TPB2)   // 50 iterations
#define CAP2 4096

// ---------------- CDNA5 async global->LDS path (guarded) ----------------
#if defined(__gfx1250__) && __has_builtin(__builtin_amdgcn_global_load_async_to_lds_b128)
#define HAVE_ASYNC_COPY 1
#else
#define HAVE_ASYNC_COPY 0
#endif

// pointer types the async builtin wants: 4 x i32 vectors in AS1 / AS3
typedef int v4i __attribute__((vector_size(16)));
typedef __attribute__((address_space(1))) v4i* gptr_v4i;
typedef __attribute__((address_space(3))) v4i* lptr_v4i;

__device__ __forceinline__ void wait_async_le(int n01) {
#if __has_builtin(__builtin_amdgcn_s_wait_asynccnt)
  if (n01 == 0) __builtin_amdgcn_s_wait_asynccnt(0);
  else          __builtin_amdgcn_s_wait_asynccnt(1);
#else
  if (n01 == 0) asm volatile("s_wait_asynccnt 0x0" ::: "memory");
  else          asm volatile("s_wait_asynccnt 0x1" ::: "memory");
#endif
}

// ---------------- key packing ----------------
// Monotonic float key in high 32 bits; ~index in low 32 bits.
// Descending u64 order == score descending, index ascending on ties
// (matches jax.lax.top_k). All keys distinct -> deterministic result.
__device__ __forceinline__ unsigned long long make_key(float v, unsigned idx) {
  unsigned b = __float_as_uint(v);
  b = (b & 0x80000000u) ? ~b : (b | 0x80000000u);
  return ((unsigned long long)b << 32) | (unsigned long long)(~idx);
}

// ---------------- block-wide bitonic sort, descending ----------------
__device__ __forceinline__ void bitonic_sort_desc(unsigned long long* buf,
                                                  unsigned cap, unsigned tid,
                                                  unsigned nt) {
  for (unsigned k = 2; k <= cap; k <<= 1) {
    for (unsigned j = k >> 1; j > 0; j >>= 1) {
      __syncthreads();
      for (unsigned i = tid; i < cap; i += nt) {
        unsigned l = i ^ j;
        if (l > i) {
          unsigned long long a = buf[i];
          unsigned long long b = buf[l];
          bool desc = ((i & k) == 0);
          if (desc ? (a < b) : (a > b)) {
            buf[i] = b;
            buf[l] = a;
          }
        }
      }
    }
  }
  __syncthreads();
}

// Sort buffer, keep top-K, raise threshold, zero the tail.
__device__ __forceinline__ void sort_and_trim(unsigned long long* buf,
                                              unsigned cap, int* cnt,
                                              unsigned long long* thr,
                                              unsigned tid, unsigned nt) {
  __syncthreads();
  bitonic_sort_desc(buf, cap, tid, nt);
  if (tid == 0) {
    *thr = buf[TOPK - 1];
    *cnt = TOPK;
  }
  for (unsigned i = TOPK + tid; i < cap; i += nt) buf[i] = 0ull;
  __syncthreads();
}

// =====================================================================
// Kernel 1: per-block exact top-100 over a contiguous 40,960-elem chunk.
// Streams via async global->LDS double buffer when available.
// =====================================================================
__global__ __launch_bounds__(TPB1) void topk_block_kernel(
    const float* __restrict__ cls, unsigned long long* __restrict__ cand) {
  __shared__ unsigned long long buf[CAP1];
  __shared__ int cnt;
  __shared__ unsigned long long thr;
#if HAVE_ASYNC_COPY
  __shared__ float4 tile[2][TPB1];
#endif

  const unsigned tid = threadIdx.x;
  for (unsigned i = tid; i < CAP1; i += TPB1) buf[i] = 0ull;
  if (tid == 0) {
    cnt = 0;
    thr = 0ull;
  }
  __syncthreads();

  const unsigned base = blockIdx.x * CHUNK;
  const float4* gsrc = reinterpret_cast<const float4*>(cls + base);

#if HAVE_ASYNC_COPY
  // prologue: stage tile 0
  __builtin_amdgcn_global_load_async_to_lds_b128(
      (gptr_v4i)(gsrc + tid), (lptr_v4i)&tile[0][tid], 0, 0);
#endif

  for (int it = 0; it < ITERS1; ++it) {
    float4 v;
#if HAVE_ASYNC_COPY
    if (it + 1 < ITERS1) {
      // stage next tile, then wait for current (<=1 outstanding)
      __builtin_amdgcn_global_load_async_to_lds_b128(
          (gptr_v4i)(gsrc + (it + 1) * TPB1 + tid),
          (lptr_v4i)&tile[(it + 1) & 1][tid], 0, 0);
      wait_async_le(1);
    } else {
      wait_async_le(0);
    }
    v = tile[it & 1][tid];  // each lane reads only bytes its own op wrote
#else
    v = gsrc[it * TPB1 + tid];
#endif
    const unsigned eidx = base + it * TILE + tid * VEC;
    const unsigned long long t = thr;
    const float* vp = &v.x;
#pragma unroll
    for (int j = 0; j < VEC; ++j) {
      unsigned long long key = make_key(vp[j], eidx + j);
      if (key > t) {
        int p = atomicAdd(&cnt, 1);  // LDS atomic (ds_add_rtn)
        buf[p] = key;
      }
    }
    __syncthreads();
    // start-of-iter invariant cnt <= CAP1-TILE; appends/iter <= TILE
    if (cnt > CAP1 - TILE) sort_and_trim(buf, CAP1, &cnt, &thr, tid, TPB1);
  }

  __syncthreads();
  bitonic_sort_desc(buf, CAP1, tid, TPB1);
  for (unsigned i = tid; i < TOPK; i += TPB1)
    cand[(size_t)blockIdx.x * TOPK + i] = buf[i];
}

// =====================================================================
// Kernel 2: merge 512*100 candidates -> global top-100, decode, gather.
// =====================================================================
__global__ __launch_bounds__(TPB2) void topk_merge_kernel(
    const unsigned long long* __restrict__ cand, const float* __restrict__ loc,
    float* __restrict__ out) {
  __shared__ unsigned long long buf[CAP2];
  __shared__ int cnt;
  __shared__ unsigned long long thr;

  const unsigned tid = threadIdx.x;
  for (unsigned i = tid; i < CAP2; i += TPB2) buf[i] = 0ull;
  if (tid == 0) {
    cnt = 0;
    thr = 0ull;
  }
  __syncthreads();

  for (int it = 0; it < ITERS2; ++it) {
    unsigned long long key = cand[(size_t)it * TPB2 + tid];
    const unsigned long long t = thr;
    if (key > t) {
      int p = atomicAdd(&cnt, 1);
      buf[p] = key;
    }
    __syncthreads();
    if (cnt > CAP2 - TPB2) sort_and_trim(buf, CAP2, &cnt, &thr, tid, TPB2);
  }

  __syncthreads();
  bitonic_sort_desc(buf, CAP2, tid, TPB2);

  if (tid < TOPK) {
    unsigned long long key = buf[tid];
    unsigned kb = (unsigned)(key >> 32);
    unsigned sb = (kb & 0x80000000u) ? (kb ^ 0x80000000u) : ~kb;
    float score = __uint_as_float(sb);
    unsigned flat = ~(unsigned)(key & 0xFFFFFFFFull);
    unsigned cls_id = flat % FC;
    unsigned spatial = flat / FC;  // == ys*FW + xs
    const float* lp = loc + (size_t)spatial * 4;
    float* o = out + (size_t)tid * 6;
    o[0] = lp[0] * BOX_SCALE;
    o[1] = lp[1] * BOX_SCALE;
    o[2] = lp[2] * BOX_SCALE;
    o[3] = lp[3] * BOX_SCALE;
    o[4] = score;
    o[5] = (float)cls_id;
  }
}

extern "C" void kernel_launch(void* const* d_in, const int* in_sizes, int n_in,
                              void* d_out, int out_size, void* d_ws,
                              size_t ws_size, hipStream_t stream) {
  (void)in_sizes;
  (void)n_in;
  (void)out_size;
  (void)ws_size;
  const float* cls = (const float*)d_in[0];  // [1,512,512,80] f32
  const float* loc = (const float*)d_in[1];  // [1,512,512,4]  f32
  float* out = (float*)d_out;                // [1,100,6] f32
  unsigned long long* cand = (unsigned long long*)d_ws;  // 512*100*8 B

  topk_block_kernel<<<B1, TPB1, 0, stream>>>(cls, cand);
  topk_merge_kernel<<<1, TPB2, 0, stream>>>(cand, loc, out);
}